// DiscreteContinuousConvS2_85847806313159
// MI455X (gfx1250) — compile-verified
//
#include <hip/hip_runtime.h>
#include <hip/hip_bf16.h>

// DISCO S2 convolution for MI455X (gfx1250), wave32 + WMMA f32 16x16x4.
//
// out[bc, k, t, p] = sum_{j in seg(k,t)} v_j * x[bc, lat_j, (lon_j - 1 - p) % 720]
//
// Per (segment, lat-ring): with w = (lon_j - lonMin) + (15 - pp), the banded
// weight matrix B[w][pp] is independent of the p-tile, and
//   out[bc, p0+pp] += sum_w A[bc,w] * B[w,pp],  A[bc,w] = x[bc, lat, (qbase+w)%720]
// which maps onto V_WMMA_F32_16X16X4_F32 (M=16 bc, N=16 p, K-loop over w).
// 8 waves/block each own a 16-bc slab; B is shared from LDS.
//
// xt is transposed AND lon-padded (rows 720..855 replicate 0..135) so the
// K-loop addressing is linear: A loads use one base VGPR + immediate offsets.

typedef __attribute__((ext_vector_type(2))) float v2f;
typedef __attribute__((ext_vector_type(8))) float v8f;

#define BC      128
#define KSZ     3
#define NLAT    361
#define NLON    720
#define LPAD    136            // >= CHW + 8 : inner-loop reach past lon 719
#define NLONP   (NLON + LPAD)  // 856 padded lon rows in xt
#define NSEG    (KSZ * NLAT)   // 1083
#define NPT     (NLON / 16)    // 45 p-tiles of 16
#define MAXNNZ  3072           // worst segment (polar) ~1440 taps
#define MAXR    16             // max distinct lat rings per segment (~8 real)
#define CHW     128            // w-chunk width for banded B (multiple of 16)

// ---------------------------------------------------------------- preprocessing
__global__ void initSeg(int* segStart, int* segEnd) {
    int s = blockIdx.x * blockDim.x + threadIdx.x;
    if (s < NSEG) { segStart[s] = 0; segEnd[s] = 0; }
}

__global__ void markSeg(const int* __restrict__ pk, const int* __restrict__ pt,
                        int nnz, int* __restrict__ segStart, int* __restrict__ segEnd) {
    int i = blockIdx.x * blockDim.x + threadIdx.x;
    if (i >= nnz) return;
    int key = pk[i] * NLAT + pt[i];
    if (i == 0) {
        segStart[key] = 0;
    } else {
        int pkey = pk[i - 1] * NLAT + pt[i - 1];
        if (pkey != key) { segStart[key] = i; segEnd[pkey] = i; }
    }
    if (i == nnz - 1) segEnd[key] = nnz;
}

// x[bc][lat][lon] -> xt[lat][lonp][bc], lon rows 720..855 replicate 0..135.
__global__ void transposeX(const float* __restrict__ x, float* __restrict__ xt) {
    __shared__ float tile[32][33];
    int la   = blockIdx.z;
    int lon0 = blockIdx.x * 32;
    int bc0  = blockIdx.y * 32;
    int tx = threadIdx.x, ty = threadIdx.y;          // block (32, 8)
    for (int i = ty; i < 32; i += 8) {
        int lon = lon0 + tx;
        tile[i][tx] = (lon < NLON) ? x[((size_t)(bc0 + i) * NLAT + la) * NLON + lon] : 0.f;
    }
    __syncthreads();
    for (int i = ty; i < 32; i += 8) {
        int lon = lon0 + i;
        if (lon < NLON) {
            float val = tile[tx][i];
            xt[((size_t)la * NLONP + lon) * BC + (bc0 + tx)] = val;
            if (lon < LPAD)
                xt[((size_t)la * NLONP + NLON + lon) * BC + (bc0 + tx)] = val;
        }
    }
}

// ---------------------------------------------------------------- main kernel
template<bool TRANSPOSED>
__global__ void __launch_bounds__(256)
discoWmma(const float* __restrict__ xsrc,
          const float* __restrict__ pv,
          const int* __restrict__ pla, const int* __restrict__ plo,
          const int* __restrict__ segStart, const int* __restrict__ segEnd,
          float* __restrict__ out) {
    __shared__ short sLa[MAXNNZ];
    __shared__ short sLo[MAXNNZ];
    __shared__ float sV [MAXNNZ];
    __shared__ float sB [CHW * 16];
    __shared__ int   sRingOff[MAXR + 1];
    __shared__ int   sNr;

    const int p0  = blockIdx.x * 16;    // p-tile
    const int s   = blockIdx.y;         // segment = k*NLAT + t
    const int kk  = s / NLAT;
    const int tt  = s % NLAT;
    const int tid = threadIdx.x;
    const int lane   = tid & 31;
    const int waveId = tid >> 5;        // 8 waves, each owns 16 bc rows
    const int bc0    = waveId * 16;
    const int nLane  = lane & 15;       // N (p) index within tile
    const int kBase  = (lane >> 4) * 2; // A/B 32-bit layout: lanes 0-15 -> K0,1; 16-31 -> K2,3
    const int bcM    = bc0 + nLane;     // M (bc) row for A fragment

    const int j0 = segStart[s];
    const int j1 = segEnd[s];
    const int n  = min(j1 - j0, MAXNNZ);

    // stage taps in LDS
    for (int i = tid; i < n; i += 256) {
        sLa[i] = (short)pla[j0 + i];
        sLo[i] = (short)plo[j0 + i];
        sV [i] = pv [j0 + i];
    }
    __syncthreads();

    // ring table (taps are sorted by (lat, lon) within a segment)
    if (tid == 0) {
        int nr = 0;
        if (n > 0) {
            sRingOff[0] = 0; nr = 1;
            short cur = sLa[0];
            for (int i = 1; i < n; ++i)
                if (sLa[i] != cur) { cur = sLa[i]; if (nr < MAXR) sRingOff[nr++] = i; }
            sRingOff[nr] = n;
        }
        sNr = nr;
    }
    __syncthreads();

    v8f acc = {};
    const int nr = sNr;

    for (int r = 0; r < nr; ++r) {
        const int ja = sRingOff[r], jb = sRingOff[r + 1];
        const int la    = sLa[ja];
        const int loMin = sLo[ja];
        const int loMax = sLo[jb - 1];
        const int W  = loMax - loMin + 16;
        const int Wp = (W + 3) & ~3;
        const int qbase = loMin - 16 - p0;   // virtual lon of w = 0

        for (int wb = 0; wb < Wp; wb += CHW) {
            const int wlen = min(CHW, Wp - wb);

            for (int i = tid; i < wlen * 16; i += 256) sB[i] = 0.f;
            __syncthreads();

            // band fill: tap j covers w in [wt, wt+15], pp = wt + 15 - w (collision-free)
            for (int j = ja + tid; j < jb; j += 256) {
                const int   wt = (int)sLo[j] - loMin;
                const float vv = sV[j];
                const int wlo = max(wt, wb);
                const int whi = min(wt + 15, wb + wlen - 1);
                for (int w = wlo; w <= whi; ++w)
                    sB[(w - wb) * 16 + (wt + 15 - w)] = vv;
            }
            __syncthreads();

            int lonStart = (qbase + wb) % NLON;
            if (lonStart < 0) lonStart += NLON;

            if (TRANSPOSED) {
                // Linear addressing into padded xt: lonStart + c + kBase + 1 <= 846 < NLONP.
                // One base address per lane half; A loads get immediate offsets (c*512B).
                const float* __restrict__ ap =
                    xsrc + ((size_t)la * NLONP + lonStart + kBase) * BC + bcM;
                const float* __restrict__ bp = sB + kBase * 16 + nLane;

                int c = 0;
                for (; c + 16 <= wlen; c += 16) {
                    #pragma unroll
                    for (int u = 0; u < 4; ++u) {
                        const int cc = c + u * 4;
                        v2f Af = { ap[(size_t)cc * BC], ap[(size_t)(cc + 1) * BC] };
                        v2f Bf = { bp[cc * 16], bp[cc * 16 + 16] };
                        acc = __builtin_amdgcn_wmma_f32_16x16x4_f32(
                                false, Af, false, Bf, (short)0, acc, false, false);
                    }
                }
                for (; c < wlen; c += 4) {
                    v2f Af = { ap[(size_t)c * BC], ap[(size_t)(c + 1) * BC] };
                    v2f Bf = { bp[c * 16], bp[c * 16 + 16] };
                    acc = __builtin_amdgcn_wmma_f32_16x16x4_f32(
                            false, Af, false, Bf, (short)0, acc, false, false);
                }
            } else {
                // Fallback: original (untransposed) x layout with explicit wrap.
                const float* __restrict__ row = xsrc + ((size_t)bcM * NLAT + la) * NLON;
                const float* __restrict__ bp  = sB + kBase * 16 + nLane;
                for (int c = 0; c < wlen; c += 4) {
                    int lonA0 = lonStart + c + kBase; lonA0 -= (lonA0 >= NLON) ? NLON : 0;
                    int lonA1 = lonA0 + 1;            lonA1 -= (lonA1 >= NLON) ? NLON : 0;
                    v2f Af = { row[lonA0], row[lonA1] };
                    v2f Bf = { bp[c * 16], bp[c * 16 + 16] };
                    acc = __builtin_amdgcn_wmma_f32_16x16x4_f32(
                            false, Af, false, Bf, (short)0, acc, false, false);
                }
            }
            __syncthreads();
        }
    }

    // C/D layout: VGPR v, lanes 0-15 -> M=v; lanes 16-31 -> M=v+8 (N = lane&15).
    const int mHalf = (lane >> 4) * 8;
    #pragma unroll
    for (int v = 0; v < 8; ++v) {
        const int bc = bc0 + v + mHalf;
        const size_t o = (((size_t)bc * KSZ + kk) * NLAT + tt) * (size_t)NLON + p0 + nLane;
        __builtin_nontemporal_store(acc[v], &out[o]);
    }
}

// ---------------------------------------------------------------- launch
extern "C" void kernel_launch(void* const* d_in, const int* in_sizes, int n_in,
                              void* d_out, int out_size, void* d_ws, size_t ws_size,
                              hipStream_t stream) {
    const float* x   = (const float*)d_in[0];
    const float* pv  = (const float*)d_in[1];
    const int*   pk  = (const int*)  d_in[2];
    const int*   pt  = (const int*)  d_in[3];
    const int*   pla = (const int*)  d_in[4];
    const int*   plo = (const int*)  d_in[5];
    const int    nnz = in_sizes[1];
    float* out = (float*)d_out;

    int* segStart = (int*)d_ws;
    int* segEnd   = segStart + NSEG;
    const size_t xtOff   = ((size_t)(2 * NSEG * sizeof(int)) + 255) & ~(size_t)255;
    const size_t xtBytes = (size_t)NLAT * NLONP * BC * sizeof(float);
    float* xt = (float*)((char*)d_ws + xtOff);
    const bool doT = (ws_size >= xtOff + xtBytes);

    initSeg<<<(NSEG + 255) / 256, 256, 0, stream>>>(segStart, segEnd);
    markSeg<<<(nnz + 255) / 256, 256, 0, stream>>>(pk, pt, nnz, segStart, segEnd);

    if (doT) {
        transposeX<<<dim3((NLON + 31) / 32, BC / 32, NLAT), dim3(32, 8), 0, stream>>>(x, xt);
        discoWmma<true><<<dim3(NPT, NSEG), 256, 0, stream>>>(xt, pv, pla, plo,
                                                             segStart, segEnd, out);
    } else {
        discoWmma<false><<<dim3(NPT, NSEG), 256, 0, stream>>>(x, pv, pla, plo,
                                                              segStart, segEnd, out);
    }
}